// E_GATLayer_70153995813481
// MI455X (gfx1250) — compile-verified
//
#include <hip/hip_runtime.h>
#include <math.h>

typedef float v2f __attribute__((ext_vector_type(2)));
typedef float v8f __attribute__((ext_vector_type(8)));

#define N_NODES 50000
#define N_EDGES 800000
#define NDIM 64          // in = edim = out = 64
#define LDS_STRIDE 65    // float2 row stride, padded to avoid bank conflicts

// -------- K0: zero accumulators (must run every call: graph replays) --------
__global__ void k_init(float* __restrict__ denom, float* __restrict__ z) {
    int i = blockIdx.x * blockDim.x + threadIdx.x;
    if (i < N_NODES * NDIM) z[i] = 0.f;
    if (i < N_NODES) denom[i] = 0.f;
}

// -------- K1: per-node attention logit parts: a_s = h·Wa[0:64], a_d = h·Wa[64:128]
// 4 lanes per node, shuffle-reduce. Tiny (12.8 MB read).
__global__ void k_node_logits(const float* __restrict__ nf, const float* __restrict__ Wa,
                              float* __restrict__ a_s, float* __restrict__ a_d) {
    int gid = blockIdx.x * blockDim.x + threadIdx.x;
    int node = gid >> 2, q = gid & 3;
    if (node >= N_NODES) return;
    const float* r = nf + (size_t)node * NDIM + q * 16;
    float ps = 0.f, pd = 0.f;
#pragma unroll
    for (int j = 0; j < 16; ++j) {
        float x = r[j];
        ps += x * Wa[q * 16 + j];
        pd += x * Wa[64 + q * 16 + j];
    }
    ps += __shfl_xor(ps, 1); ps += __shfl_xor(ps, 2);
    pd += __shfl_xor(pd, 1); pd += __shfl_xor(pd, 2);
    if (q == 0) { a_s[node] = ps; a_d[node] = pd; }
}

// -------- K2: fused edge pass --------------------------------------------
// Per wave: 16 edges. A-tile = [h_src | efeat] (16x128 fp32), B = W_ne in LDS.
// m = A@W_ne via 128x v_wmma_f32_16x16x4_f32; attention dot folded into the
// same efeat fragment loads; exp(e) accumulated into denom; exp(e)*m
// scattered into z via f32 atomics (unnormalized softmax; divide in K3).
__global__ void __launch_bounds__(256) k_edge(
    const float* __restrict__ nf, const float* __restrict__ ef,
    const float* __restrict__ Wne, const float* __restrict__ Wattn,
    const float* __restrict__ battn,
    const int* __restrict__ src, const int* __restrict__ dst,
    const float* __restrict__ a_s, const float* __restrict__ a_d,
    float* __restrict__ denom, float* __restrict__ z) {

    __shared__ float2 sW[64 * LDS_STRIDE];  // sW[h][c] = {W[2h][c], W[2h+1][c]}
    __shared__ float  sWa[64];              // W_attn[128:192] (efeat part)

    int tid = threadIdx.x;
    for (int i = tid; i < 64 * 64; i += 256) {
        int h = i >> 6, c = i & 63;
        sW[h * LDS_STRIDE + c] = make_float2(Wne[(2 * h) * NDIM + c],
                                             Wne[(2 * h + 1) * NDIM + c]);
    }
    if (tid < 64) sWa[tid] = Wattn[128 + tid];
    __syncthreads();

    int wave = tid >> 5, lane = tid & 31;
    int row = lane & 15, hi = lane >> 4, kb = hi * 2;

    int tile = blockIdx.x * 8 + wave;       // N_EDGES/16 tiles, exact multiple
    int e0 = tile * 16;
    int erow = e0 + row;
    int s = src[erow];
    int d = dst[erow];

    const float2* hs = (const float2*)(nf + (size_t)s * NDIM);
    const float2* ev = (const float2*)(ef + (size_t)erow * NDIM);

    v8f acc0 = {}, acc1 = {}, acc2 = {}, acc3 = {};
    float p = 0.f;  // partial attention dot over efeat

    // first 64 K: h_src
#pragma unroll
    for (int kt = 0; kt < 16; ++kt) {
        int hb = kt * 2 + hi;               // (k0+kb)/2
        float2 av = hs[hb];
        v2f a = {av.x, av.y};
        float2 t;  v2f b;
        t = sW[hb * LDS_STRIDE +  0 + row]; b.x = t.x; b.y = t.y;
        acc0 = __builtin_amdgcn_wmma_f32_16x16x4_f32(false, a, false, b, (short)0, acc0, false, false);
        t = sW[hb * LDS_STRIDE + 16 + row]; b.x = t.x; b.y = t.y;
        acc1 = __builtin_amdgcn_wmma_f32_16x16x4_f32(false, a, false, b, (short)0, acc1, false, false);
        t = sW[hb * LDS_STRIDE + 32 + row]; b.x = t.x; b.y = t.y;
        acc2 = __builtin_amdgcn_wmma_f32_16x16x4_f32(false, a, false, b, (short)0, acc2, false, false);
        t = sW[hb * LDS_STRIDE + 48 + row]; b.x = t.x; b.y = t.y;
        acc3 = __builtin_amdgcn_wmma_f32_16x16x4_f32(false, a, false, b, (short)0, acc3, false, false);
    }
    // second 64 K: efeat (also feeds the attention dot)
#pragma unroll
    for (int kt = 0; kt < 16; ++kt) {
        int kk = kt * 4 + kb;
        float2 av = ev[kt * 2 + hi];
        p += av.x * sWa[kk] + av.y * sWa[kk + 1];
        v2f a = {av.x, av.y};
        int hb = 32 + kt * 2 + hi;
        float2 t;  v2f b;
        t = sW[hb * LDS_STRIDE +  0 + row]; b.x = t.x; b.y = t.y;
        acc0 = __builtin_amdgcn_wmma_f32_16x16x4_f32(false, a, false, b, (short)0, acc0, false, false);
        t = sW[hb * LDS_STRIDE + 16 + row]; b.x = t.x; b.y = t.y;
        acc1 = __builtin_amdgcn_wmma_f32_16x16x4_f32(false, a, false, b, (short)0, acc1, false, false);
        t = sW[hb * LDS_STRIDE + 32 + row]; b.x = t.x; b.y = t.y;
        acc2 = __builtin_amdgcn_wmma_f32_16x16x4_f32(false, a, false, b, (short)0, acc2, false, false);
        t = sW[hb * LDS_STRIDE + 48 + row]; b.x = t.x; b.y = t.y;
        acc3 = __builtin_amdgcn_wmma_f32_16x16x4_f32(false, a, false, b, (short)0, acc3, false, false);
    }

    // attention logit -> exp weight (softmax is shift-invariant; normalize later)
    p += __shfl_xor(p, 16);                 // combine even/odd K halves
    float el = a_s[s] + a_d[d] + p + battn[0];
    el = el > 0.f ? el : 0.01f * el;        // leaky_relu, slope 0.01
    float ex = __expf(el);
    if (hi == 0) atomicAdd(&denom[d], ex);

    // scatter exp(e)*m into z[dst]
#pragma unroll
    for (int r = 0; r < 8; ++r) {
        int rr = r + 8 * hi;                // output row held by this VGPR
        float exr = __shfl(ex, rr);
        int   dr  = __shfl(d, rr);
        float* zr = z + (size_t)dr * NDIM + row;
        atomicAdd(zr +  0, exr * acc0[r]);
        atomicAdd(zr + 16, exr * acc1[r]);
        atomicAdd(zr + 32, exr * acc2[r]);
        atomicAdd(zr + 48, exr * acc3[r]);
    }
}

// -------- K3: per-node output: out = relu([h, z/denom] @ W_out) ----------
__global__ void __launch_bounds__(256) k_out(
    const float* __restrict__ nf, const float* __restrict__ z,
    const float* __restrict__ denom, const float* __restrict__ Wout,
    float* __restrict__ out) {

    __shared__ float2 sW[64 * LDS_STRIDE];
    int tid = threadIdx.x;
    for (int i = tid; i < 64 * 64; i += 256) {
        int h = i >> 6, c = i & 63;
        sW[h * LDS_STRIDE + c] = make_float2(Wout[(2 * h) * NDIM + c],
                                             Wout[(2 * h + 1) * NDIM + c]);
    }
    __syncthreads();

    int wave = tid >> 5, lane = tid & 31;
    int row = lane & 15, hi = lane >> 4;
    int tile = blockIdx.x * 8 + wave;
    if (tile >= N_NODES / 16) return;       // uniform per-wave branch: EXEC stays full

    int n0 = tile * 16;
    int n = n0 + row;
    const float2* xf = (const float2*)(nf + (size_t)n * NDIM);
    const float2* zf = (const float2*)(z + (size_t)n * NDIM);
    float dn = denom[n];
    float invd = dn > 0.f ? 1.0f / dn : 0.f;  // empty nodes -> z row = 0 (matches ref)

    v8f acc0 = {}, acc1 = {}, acc2 = {}, acc3 = {};
#pragma unroll
    for (int kt = 0; kt < 16; ++kt) {
        int hb = kt * 2 + hi;
        float2 av = xf[hb];
        v2f a = {av.x, av.y};
        float2 t;  v2f b;
        t = sW[hb * LDS_STRIDE +  0 + row]; b.x = t.x; b.y = t.y;
        acc0 = __builtin_amdgcn_wmma_f32_16x16x4_f32(false, a, false, b, (short)0, acc0, false, false);
        t = sW[hb * LDS_STRIDE + 16 + row]; b.x = t.x; b.y = t.y;
        acc1 = __builtin_amdgcn_wmma_f32_16x16x4_f32(false, a, false, b, (short)0, acc1, false, false);
        t = sW[hb * LDS_STRIDE + 32 + row]; b.x = t.x; b.y = t.y;
        acc2 = __builtin_amdgcn_wmma_f32_16x16x4_f32(false, a, false, b, (short)0, acc2, false, false);
        t = sW[hb * LDS_STRIDE + 48 + row]; b.x = t.x; b.y = t.y;
        acc3 = __builtin_amdgcn_wmma_f32_16x16x4_f32(false, a, false, b, (short)0, acc3, false, false);
    }
#pragma unroll
    for (int kt = 0; kt < 16; ++kt) {
        float2 av = zf[kt * 2 + hi];
        av.x *= invd; av.y *= invd;
        v2f a = {av.x, av.y};
        int hb = 32 + kt * 2 + hi;
        float2 t;  v2f b;
        t = sW[hb * LDS_STRIDE +  0 + row]; b.x = t.x; b.y = t.y;
        acc0 = __builtin_amdgcn_wmma_f32_16x16x4_f32(false, a, false, b, (short)0, acc0, false, false);
        t = sW[hb * LDS_STRIDE + 16 + row]; b.x = t.x; b.y = t.y;
        acc1 = __builtin_amdgcn_wmma_f32_16x16x4_f32(false, a, false, b, (short)0, acc1, false, false);
        t = sW[hb * LDS_STRIDE + 32 + row]; b.x = t.x; b.y = t.y;
        acc2 = __builtin_amdgcn_wmma_f32_16x16x4_f32(false, a, false, b, (short)0, acc2, false, false);
        t = sW[hb * LDS_STRIDE + 48 + row]; b.x = t.x; b.y = t.y;
        acc3 = __builtin_amdgcn_wmma_f32_16x16x4_f32(false, a, false, b, (short)0, acc3, false, false);
    }

#pragma unroll
    for (int r = 0; r < 8; ++r) {
        int nr = n0 + r + 8 * hi;
        float* o = out + (size_t)nr * NDIM + row;
        o[ 0] = fmaxf(acc0[r], 0.f);
        o[16] = fmaxf(acc1[r], 0.f);
        o[32] = fmaxf(acc2[r], 0.f);
        o[48] = fmaxf(acc3[r], 0.f);
    }
}

extern "C" void kernel_launch(void* const* d_in, const int* in_sizes, int n_in,
                              void* d_out, int out_size, void* d_ws, size_t ws_size,
                              hipStream_t stream) {
    const float* nf    = (const float*)d_in[0];
    const float* ef    = (const float*)d_in[1];
    const float* Wne   = (const float*)d_in[2];
    const float* Wattn = (const float*)d_in[3];
    const float* battn = (const float*)d_in[4];
    const float* Wout  = (const float*)d_in[5];
    const int*   src   = (const int*)d_in[6];
    const int*   dst   = (const int*)d_in[7];
    float* out = (float*)d_out;

    float* wsf   = (float*)d_ws;
    float* a_s   = wsf;
    float* a_d   = wsf + N_NODES;
    float* denom = wsf + 2 * N_NODES;
    float* z     = wsf + 3 * N_NODES;      // N_NODES * 64 floats

    k_init<<<(N_NODES * NDIM + 255) / 256, 256, 0, stream>>>(denom, z);
    k_node_logits<<<(N_NODES * 4 + 255) / 256, 256, 0, stream>>>(nf, Wattn, a_s, a_d);
    k_edge<<<N_EDGES / 128, 256, 0, stream>>>(nf, ef, Wne, Wattn, battn, src, dst,
                                              a_s, a_d, denom, z);
    k_out<<<(N_NODES / 16 + 7) / 8, 256, 0, stream>>>(nf, z, denom, Wout, out);
}